// KinematicLayer_3058016714901
// MI455X (gfx1250) — compile-verified
//
#include <hip/hip_runtime.h>

#define BS   256
#define NIN  26
#define NOUT 51

typedef float v4f __attribute__((ext_vector_type(4)));
typedef float v2f __attribute__((ext_vector_type(2)));
typedef int   v4i __attribute__((ext_vector_type(4)));

#if defined(__has_builtin)
#if __has_builtin(__builtin_amdgcn_global_load_async_to_lds_b32) && \
    __has_builtin(__builtin_amdgcn_s_wait_asynccnt)
#define USE_ASYNC_LDS 1
#endif
#endif
#ifndef USE_ASYNC_LDS
#define USE_ASYNC_LDS 0
#endif

#if USE_ASYNC_LDS && __has_builtin(__builtin_amdgcn_global_load_async_to_lds_b128)
#define HAVE_ASYNC_B128 1
#else
#define HAVE_ASYNC_B128 0
#endif

#if USE_ASYNC_LDS
typedef __attribute__((address_space(1))) int gas_int;
typedef __attribute__((address_space(3))) int las_int;
typedef __attribute__((address_space(1))) v4i gas_v4i;
typedef __attribute__((address_space(3))) v4i las_v4i;
#endif

// ---- post-multiply helpers: R <- R @ Rot_axis(a); t <- t + d * col(R) ----
__device__ __forceinline__ void rotZ(float r[9], float a) {
  float s = __sinf(a), c = __cosf(a);
#pragma unroll
  for (int i = 0; i < 3; ++i) {
    float x = r[3*i+0], y = r[3*i+1];
    r[3*i+0] = fmaf(c, x,  s * y);
    r[3*i+1] = fmaf(c, y, -s * x);
  }
}
__device__ __forceinline__ void rotX(float r[9], float a) {
  float s = __sinf(a), c = __cosf(a);
#pragma unroll
  for (int i = 0; i < 3; ++i) {
    float y = r[3*i+1], z = r[3*i+2];
    r[3*i+1] = fmaf(c, y,  s * z);
    r[3*i+2] = fmaf(c, z, -s * y);
  }
}
__device__ __forceinline__ void rotY(float r[9], float a) {
  float s = __sinf(a), c = __cosf(a);
#pragma unroll
  for (int i = 0; i < 3; ++i) {
    float x = r[3*i+0], z = r[3*i+2];
    r[3*i+0] = fmaf(c, x, -s * z);
    r[3*i+2] = fmaf(c, z,  s * x);
  }
}
__device__ __forceinline__ void transY(const float r[9], float t[3], float d) {
#pragma unroll
  for (int i = 0; i < 3; ++i) t[i] = fmaf(d, r[3*i+1], t[i]);
}

__global__ __launch_bounds__(BS)
void kin_kernel(const float* __restrict__ x, float* __restrict__ out, int n) {
  // One buffer, reused: input stage needs BS*26 floats, output stage BS*51.
  __shared__ __align__(16) float smem[BS * NOUT];  // 52224 bytes

  const int tid    = threadIdx.x;
  const int base   = (int)blockIdx.x * BS;     // first sample of this block
  const int inBase = base * NIN;
  const bool full  = (base + BS) <= n;         // true for every block when n%BS==0

  // ---- Phase 1: coalesced global -> LDS staging (async path on gfx1250) ----
  if (full) {
    const float* gin = x + inBase;             // 16B-aligned: base*NIN*4 = blk*26624
#if HAVE_ASYNC_B128
#pragma unroll
    for (int i = 0; i < 6; ++i) {              // 6*256 float4 = 6144 floats
      int idx4 = i * BS + tid;
      __builtin_amdgcn_global_load_async_to_lds_b128(
          (gas_v4i*)(gin + 4 * idx4), (las_v4i*)(&smem[4 * idx4]), 0, 0);
    }
#pragma unroll
    for (int i = 0; i < 2; ++i) {              // remaining 512 floats
      int idx = 6144 + i * BS + tid;
      __builtin_amdgcn_global_load_async_to_lds_b32(
          (gas_int*)(gin + idx), (las_int*)(&smem[idx]), 0, 0);
    }
#elif USE_ASYNC_LDS
#pragma unroll
    for (int i = 0; i < NIN; ++i) {
      int idx = i * BS + tid;
      __builtin_amdgcn_global_load_async_to_lds_b32(
          (gas_int*)(gin + idx), (las_int*)(&smem[idx]), 0, 0);
    }
#else
#pragma unroll
    for (int i = 0; i < 6; ++i) {
      int idx4 = i * BS + tid;
      *(v4f*)&smem[4 * idx4] = ((const v4f*)gin)[idx4];
    }
#pragma unroll
    for (int i = 0; i < 2; ++i) {
      int idx = 6144 + i * BS + tid;
      smem[idx] = gin[idx];
    }
#endif
  } else {
    const int inTot = n * NIN;
#pragma unroll 1
    for (int i = 0; i < NIN; ++i) {
      int idx = i * BS + tid;
      int g   = inBase + idx;
      if (g < inTot) smem[idx] = x[g];
    }
  }
#if USE_ASYNC_LDS
  __builtin_amdgcn_s_wait_asynccnt(0);
#endif
  __syncthreads();

  // ---- Phase 2: pull this thread's row into registers (13x b64 LDS reads) ----
  const int sample = base + tid;
  const bool active = sample < n;
  float m[NIN];
  if (active) {
    const v2f* r2 = (const v2f*)&smem[tid * NIN];  // 104B row stride -> 8B aligned
#pragma unroll
    for (int k = 0; k < NIN / 2; ++k) {
      v2f v = r2[k];
      m[2*k]   = v.x;
      m[2*k+1] = v.y;
    }
  }
  __syncthreads();  // all rows consumed; smem can be overwritten with outputs

  if (active) {
    float* row = &smem[tid * NOUT];
    const float scale = m[25];
    const float dAnk  = (300.0f/300.0f) * scale;  // KNEE2ANKLE
    const float dKnee = (350.0f/300.0f) * scale;  // HIP2KNEE
    const float dHip  = ( 75.0f/300.0f) * scale;  // PELVIS2HIP
    const float dTor  = (400.0f/300.0f) * scale;  // PELVIS2TORSO
    const float dNeck = ( 73.96f/300.0f) * scale; // TORSO2NECK
    const float dHead = (249.03f/300.0f) * scale; // NECK2HEAD
    const float dWr   = (250.0f/300.0f) * scale;  // ELBOW2WRIST
    const float dElb  = (250.0f/300.0f) * scale;  // SHOULDER2ELBOW
    const float dSh   = (170.0f/300.0f) * scale;  // TORSO2SHOULDER

    // pelvis: Rz(0) Rx(1) Ry(2); translation = 0
    float pel[9] = {1.f,0.f,0.f, 0.f,1.f,0.f, 0.f,0.f,1.f};
    rotZ(pel, m[0]); rotX(pel, m[1]); rotY(pel, m[2]);
    row[0] = 0.f; row[1] = 0.f; row[2] = 0.f;          // p0 pelvis

    float cur[9], t[3];
    // ---- left leg (uses pelvis R) ----
#pragma unroll
    for (int i = 0; i < 9; ++i) cur[i] = pel[i];
    t[0] = dHip * pel[0]; t[1] = dHip * pel[3]; t[2] = dHip * pel[6];
    row[12] = t[0]; row[13] = t[1]; row[14] = t[2];    // p4 lhp
    rotZ(cur, m[9]); rotX(cur, m[10]); rotY(cur, m[11]);
    transY(cur, t, -dKnee);
    row[15] = t[0]; row[16] = t[1]; row[17] = t[2];    // p5 lkn
    rotX(cur, m[12]);
    transY(cur, t, -dAnk);
    row[18] = t[0]; row[19] = t[1]; row[20] = t[2];    // p6 lan
    const float lan0 = t[0], lan1 = t[1], lan2 = t[2];

    // ---- right leg ----
#pragma unroll
    for (int i = 0; i < 9; ++i) cur[i] = pel[i];
    t[0] = -dHip * pel[0]; t[1] = -dHip * pel[3]; t[2] = -dHip * pel[6];
    row[21] = t[0]; row[22] = t[1]; row[23] = t[2];    // p7 rhp
    rotZ(cur, m[13]); rotX(cur, m[14]); rotY(cur, m[15]);
    transY(cur, t, -dKnee);
    row[24] = t[0]; row[25] = t[1]; row[26] = t[2];    // p8 rkn
    const float rkn0 = t[0], rkn1 = t[1], rkn2 = t[2];
    rotX(cur, m[16]);
    transY(cur, t, -dAnk);
    row[27] = t[0]; row[28] = t[1]; row[29] = t[2];    // p9 ran

    // ---- torso (overwrite pel in place: pel now holds torso R) ----
    rotZ(pel, m[3]); rotY(pel, m[4]);
    float ttor[3];
    ttor[0] = dTor * pel[1]; ttor[1] = dTor * pel[4]; ttor[2] = dTor * pel[7];
    row[3] = ttor[0]; row[4] = ttor[1]; row[5] = ttor[2];  // p1 torso

    // ---- neck + head ----
#pragma unroll
    for (int i = 0; i < 9; ++i) cur[i] = pel[i];
    t[0] = ttor[0]; t[1] = ttor[1]; t[2] = ttor[2];
    rotZ(cur, m[5]); rotX(cur, m[6]); rotY(cur, m[7]);
    transY(cur, t, dNeck);
    row[6] = t[0]; row[7] = t[1]; row[8] = t[2];       // p2 neck
    rotX(cur, m[8]);
    transY(cur, t, dHead);
    row[9] = t[0]; row[10] = t[1]; row[11] = t[2];     // p3 head

    // ---- left arm ----
#pragma unroll
    for (int i = 0; i < 9; ++i) cur[i] = pel[i];
    t[0] = fmaf(dSh, pel[0], ttor[0]);
    t[1] = fmaf(dSh, pel[3], ttor[1]);
    t[2] = fmaf(dSh, pel[6], ttor[2]);
    row[30] = t[0]; row[31] = t[1]; row[32] = t[2];    // p10 lsh
    rotZ(cur, m[17]); rotX(cur, m[18]); rotY(cur, m[19]);
    transY(cur, t, -dElb);
    row[33] = t[0]; row[34] = t[1]; row[35] = t[2];    // p11 lel
    rotX(cur, m[20]);
    transY(cur, t, -dWr);
    row[36] = t[0]; row[37] = t[1]; row[38] = t[2];    // p12 lwr

    // ---- right arm ----
#pragma unroll
    for (int i = 0; i < 9; ++i) cur[i] = pel[i];
    t[0] = fmaf(-dSh, pel[0], ttor[0]);
    t[1] = fmaf(-dSh, pel[3], ttor[1]);
    t[2] = fmaf(-dSh, pel[6], ttor[2]);
    row[39] = t[0]; row[40] = t[1]; row[41] = t[2];    // p13 rsh
    rotZ(cur, m[21]); rotX(cur, m[22]); rotY(cur, m[23]);
    transY(cur, t, -dElb);
    row[42] = t[0]; row[43] = t[1]; row[44] = t[2];    // p14 rel
    rotX(cur, m[24]);
    transY(cur, t, -dWr);
    row[45] = t[0]; row[46] = t[1]; row[47] = t[2];    // p15 rwr

    // thorax = 0.5*(p_rkn + p_lan)
    row[48] = 0.5f * (rkn0 + lan0);
    row[49] = 0.5f * (rkn1 + lan1);
    row[50] = 0.5f * (rkn2 + lan2);
  }
  __syncthreads();

  // ---- Phase 3: coalesced non-temporal stores (b128 + b32 tail) ----
  const int outBase = base * NOUT;               // 16B-aligned: blk*52224 bytes
  if (full) {
    v4f* o4 = (v4f*)(out + outBase);
    const v4f* s4 = (const v4f*)smem;
#pragma unroll
    for (int i = 0; i < 12; ++i) {               // 12*256 float4 = 12288 floats
      int idx4 = i * BS + tid;
      __builtin_nontemporal_store(s4[idx4], &o4[idx4]);
    }
#pragma unroll
    for (int i = 0; i < 3; ++i) {                // remaining 768 floats
      int idx = 12288 + i * BS + tid;
      __builtin_nontemporal_store(smem[idx], &out[outBase + idx]);
    }
  } else {
    const int outTot = n * NOUT;
#pragma unroll 1
    for (int i = 0; i < NOUT; ++i) {
      int idx = i * BS + tid;
      int g   = outBase + idx;
      if (g < outTot) __builtin_nontemporal_store(smem[idx], &out[g]);
    }
  }
}

extern "C" void kernel_launch(void* const* d_in, const int* in_sizes, int n_in,
                              void* d_out, int out_size, void* d_ws, size_t ws_size,
                              hipStream_t stream) {
  (void)n_in; (void)d_ws; (void)ws_size; (void)out_size;
  const float* x = (const float*)d_in[0];
  float* out = (float*)d_out;
  const int n = in_sizes[0] / NIN;          // 524288
  const int blocks = (n + BS - 1) / BS;     // 2048
  kin_kernel<<<blocks, BS, 0, stream>>>(x, out, n);
}